// DEKRPoseEstimationDecodeCallback_16389595201606
// MI455X (gfx1250) — compile-verified
//
#include <hip/hip_runtime.h>
#include <hip/hip_bf16.h>

typedef __attribute__((ext_vector_type(2))) float v2f;
typedef __attribute__((ext_vector_type(8))) float v8f;

#define NCH   18
#define NJ    17
#define MAXP  30
#define OFF_POSES 0
#define OFF_MEANS 1530
#define OFF_FINAL 1560
#define OFF_HSUM  1590

// ---------------------------------------------------------------------------
// Kernel 1: bilinear x4 upsample (18,256,256) -> (18,1024,1024), edge-padded.
// linspace(0,255,1021) has exact step 0.25, and rows/cols 1021..1023 replicate
// index 1020 (== source 255). So output (r,s) samples (min(r,1020)*0.25,
// min(s,1020)*0.25). Each thread produces a 4x4 patch from 4 source scalars.
// Memory-bound: 75.5 MB of writes -> coalesced float2 (b64) stores (base is
// only 8B-aligned since heatmap_sum starts at d_out element 1590).
// ---------------------------------------------------------------------------
__global__ __launch_bounds__(256) void dekr_upsample(const float* __restrict__ heat,
                                                     float* __restrict__ hsum) {
  int t  = blockIdx.x * 256 + threadIdx.x;   // 18*256*256 threads
  int q  = t & 255;                          // source col
  int rr = (t >> 8) & 255;                   // source row
  int c  = t >> 16;                          // channel
  if (c >= NCH) return;
  const float* src = heat + (size_t)c * 65536;
  int x1 = q  < 255 ? q  + 1 : 255;
  int y1 = rr < 255 ? rr + 1 : 255;
  float a00 = src[rr * 256 + q];
  float a01 = src[rr * 256 + x1];
  float a10 = src[y1 * 256 + q];
  float a11 = src[y1 * 256 + x1];
  float* dst = hsum + (size_t)c * (1024u * 1024u) + (size_t)(rr * 4) * 1024 + q * 4;
#pragma unroll
  for (int k = 0; k < 4; ++k) {
    float wy = 0.25f * (float)k;             // exact; irrelevant when rr==255
    float t0 = a00 + wy * (a10 - a00);
    float t1 = a01 + wy * (a11 - a01);
    float d  = t1 - t0;
    float2 lo, hi;
    lo.x = t0;              lo.y = t0 + 0.25f * d;
    hi.x = t0 + 0.50f * d;  hi.y = t0 + 0.75f * d;
    float* row = dst + (size_t)k * 1024;
    *reinterpret_cast<float2*>(row)     = lo;
    *reinterpret_cast<float2*>(row + 2) = hi;
  }
}

// ---------------------------------------------------------------------------
// Kernel 2: 5x5 local-max peak map of the center channel (map_size=256 -> k=5)
// into workspace (65536 floats). Source is 256KB -> L2 resident.
// ---------------------------------------------------------------------------
__global__ __launch_bounds__(256) void dekr_peak(const float* __restrict__ heat,
                                                 float* __restrict__ peak) {
  int t = blockIdx.x * 256 + threadIdx.x;    // 65536 threads
  int x = t & 255, y = t >> 8;
  const float* ctr = heat + (size_t)(NCH - 1) * 65536;
  float v = ctr[t];
  float m = -__builtin_inff();
#pragma unroll
  for (int dy = -2; dy <= 2; ++dy) {
    int yy = y + dy;
    if (yy < 0 || yy > 255) continue;
#pragma unroll
    for (int dx = -2; dx <= 2; ++dx) {
      int xx = x + dx;
      if (xx < 0 || xx > 255) continue;
      float n = ctr[yy * 256 + xx];
      m = n > m ? n : m;
    }
  }
  peak[t] = (m == v) ? v : 0.0f;
}

// ---------------------------------------------------------------------------
// Kernel 3: full decode in one 1024-thread workgroup (32 wave32 waves).
//  - iterative top-30 (stable lowest-index tie-break == jax top_k)
//  - pose coords + heatvals
//  - pairwise squared distances via V_WMMA_F32_16X16X4_F32 on wave 0:
//      A row m = [x_m, y_m, |p_m|^2, 1], B col n = [-2x_n, -2y_n, 1, |p_n|^2]
//      => D[m][n] = (x_m-x_n)^2 + (y_m-y_n)^2  (dist^2, f32 exact path)
//    "dist < thre" <=> "dist^2 < thre^2" since both are >= 0 (no sqrt).
//  - sequential NMS keep-loop on thread 0, masked output writes.
// ---------------------------------------------------------------------------
__global__ __launch_bounds__(1024) void dekr_decode(
    const float* __restrict__ offs, const float* __restrict__ peak,
    const float* __restrict__ hsum, float* __restrict__ out_poses,
    float* __restrict__ out_means, float* __restrict__ out_final) {
  __shared__ float s_rv[32];
  __shared__ int   s_ri[32];
  __shared__ float s_score[MAXP];
  __shared__ int   s_pos[MAXP];
  __shared__ float s_px[MAXP * NJ], s_py[MAXP * NJ], s_hv[MAXP * NJ];
  __shared__ float s_hs[32];
  __shared__ float s_th2[32];
  __shared__ int   s_valid[32];
  __shared__ unsigned char s_nms[32 * 32];
  __shared__ float s_mask[MAXP];

  const int t    = threadIdx.x;
  const int lane = t & 31;
  const int wid  = t >> 5;

  // ---- iterative top-30 over 65536 peak values ----
  unsigned long long excl = 0ull;
  for (int sel = 0; sel < MAXP; ++sel) {
    float bv = -__builtin_inff();
    int   bi = 0x7fffffff;
    for (int i = 0; i < 64; ++i) {
      int idx = t + (i << 10);
      float v = ((excl >> i) & 1ull) ? -__builtin_inff() : peak[idx];
      if (v > bv || (v == bv && idx < bi)) { bv = v; bi = idx; }
    }
    for (int o = 16; o > 0; o >>= 1) {               // wave32 reduction
      float ov = __shfl_down(bv, o, 32);
      int   oi = __shfl_down(bi, o, 32);
      if (ov > bv || (ov == bv && oi < bi)) { bv = ov; bi = oi; }
    }
    if (lane == 0) { s_rv[wid] = bv; s_ri[wid] = bi; }
    __syncthreads();
    if (wid == 0) {
      bv = s_rv[lane]; bi = s_ri[lane];
      for (int o = 16; o > 0; o >>= 1) {
        float ov = __shfl_down(bv, o, 32);
        int   oi = __shfl_down(bi, o, 32);
        if (ov > bv || (ov == bv && oi < bi)) { bv = ov; bi = oi; }
      }
      if (lane == 0) { s_score[sel] = bv; s_pos[sel] = bi; }
    }
    __syncthreads();
    int w = s_pos[sel];
    if ((w & 1023) == t) excl |= 1ull << (w >> 10);  // owner excludes winner
  }

  // ---- pose coords (stride 4) + heatvals from upsampled map ----
  if (t < MAXP * NJ) {
    int p = t / NJ, j = t - p * NJ;
    int pos = s_pos[p];
    float lx = (float)(pos & 255);
    float ly = (float)(pos >> 8);
    float px = 4.0f * (lx - offs[(size_t)(2 * j)     * 65536 + pos]);
    float py = 4.0f * (ly - offs[(size_t)(2 * j + 1) * 65536 + pos]);
    s_px[t] = px; s_py[t] = py;
    int xb = (int)floorf(px); xb = xb < 0 ? 0 : (xb > 1023 ? 1023 : xb);
    int yb = (int)floorf(py); yb = yb < 0 ? 0 : (yb > 1023 ? 1023 : yb);
    s_hv[t] = hsum[(size_t)j * (1024u * 1024u) + (size_t)yb * 1024 + xb];
  }
  __syncthreads();

  // ---- per-person stats: heat_score mean, area -> thre^2, validity ----
  if (t < 32) {
    if (t < MAXP) {
      float sum = 0.f;
      float mnx =  __builtin_inff(), mxx = -__builtin_inff();
      float mny =  __builtin_inff(), mxy = -__builtin_inff();
      for (int j = 0; j < NJ; ++j) {
        float hv = s_hv[t * NJ + j]; sum += hv;
        float x = s_px[t * NJ + j], y = s_py[t * NJ + j];
        mnx = x < mnx ? x : mnx; mxx = x > mxx ? x : mxx;
        mny = y < mny ? y : mny; mxy = y > mxy ? y : mxy;
      }
      s_hs[t] = sum * (1.0f / NJ);
      float we = mxx - mnx, he = mxy - mny;
      s_th2[t] = 0.0025f * (we * we + he * he);      // (NMS_THR^2) * area
      s_valid[t] = (s_score[t] > 0.01f) ? 1 : 0;
    } else {
      s_hs[t] = 0.f; s_th2[t] = 0.f; s_valid[t] = 0;
    }
  }
  __syncthreads();

  // ---- pairwise dist^2 via WMMA f32 16x16x4 on wave 0 (EXEC all ones) ----
  if (t < 32) {
    const int half = lane >> 4;                 // A/B K-halves per ISA layout
    const int ml   = lane & 15;
    float th2r[2][8];
#pragma unroll
    for (int ti = 0; ti < 2; ++ti)
#pragma unroll
      for (int v = 0; v < 8; ++v) th2r[ti][v] = s_th2[ti * 16 + v + 8 * half];
    int cnt[4][8];
#pragma unroll
    for (int pq = 0; pq < 4; ++pq)
#pragma unroll
      for (int v = 0; v < 8; ++v) cnt[pq][v] = 0;

    for (int jj = 0; jj < NJ; ++jj) {
      v2f a[2], b[2];
#pragma unroll
      for (int ttile = 0; ttile < 2; ++ttile) {
        int ia = ttile * 16 + ml;
        float ax = (ia < MAXP) ? s_px[ia * NJ + jj] : 0.f;
        float ay = (ia < MAXP) ? s_py[ia * NJ + jj] : 0.f;
        float nn = ax * ax + ay * ay;
        v2f av, bb;
        if (half == 0) { av.x = ax; av.y = ay; bb.x = -2.f * ax; bb.y = -2.f * ay; }
        else           { av.x = nn; av.y = 1.f; bb.x = 1.f;      bb.y = nn;        }
        a[ttile] = av; b[ttile] = bb;
      }
#pragma unroll
      for (int pq = 0; pq < 4; ++pq) {
        v8f c = {};
        v8f d = __builtin_amdgcn_wmma_f32_16x16x4_f32(
            /*neg_a=*/false, a[pq >> 1], /*neg_b=*/false, b[pq & 1],
            /*c_mod=*/(short)0, c, /*reuse_a=*/false, /*reuse_b=*/false);
#pragma unroll
        for (int v = 0; v < 8; ++v)
          cnt[pq][v] += (d[v] < th2r[pq >> 1][v]) ? 1 : 0;
      }
    }
#pragma unroll
    for (int pq = 0; pq < 4; ++pq)
#pragma unroll
      for (int v = 0; v < 8; ++v) {
        int i = (pq >> 1) * 16 + v + 8 * half;   // C/D layout: M = v + 8*(lane/16)
        int j = (pq & 1) * 16 + ml;              // N = lane % 16
        s_nms[i * 32 + j] =
            (cnt[pq][v] > 8 && s_valid[i] && s_valid[j]) ? 1 : 0;
      }
  }
  __syncthreads();

  // ---- sequential NMS keep-loop (exact fori_loop semantics) ----
  if (t == 0) {
    bool ignored[MAXP], keep[MAXP];
    for (int i = 0; i < MAXP; ++i) { ignored[i] = !s_valid[i]; keep[i] = false; }
    for (int i = 0; i < MAXP; ++i) {
      const unsigned char* row = &s_nms[i * 32];
      bool any = false;
      float best = -__builtin_inff();
      int ki = 0;                                 // argmax of all -inf -> 0
      for (int j = 0; j < MAXP; ++j) {
        if (row[j]) {
          any = true;
          if (s_hs[j] > best) { best = s_hs[j]; ki = j; }
        }
      }
      bool cond = !ignored[i] && any && !ignored[ki];
      if (cond) {
        keep[ki] = true;
        for (int j = 0; j < MAXP; ++j) ignored[j] = ignored[j] || (row[j] != 0);
      }
    }
    for (int p = 0; p < MAXP; ++p) s_mask[p] = keep[p] ? 1.f : 0.f;
  }
  __syncthreads();

  // ---- masked outputs ----
  if (t < MAXP) {
    float ms  = s_score[t] * s_hs[t];             // pose_score.mean(axis=1)
    bool fin  = (s_mask[t] != 0.f) && (ms >= 0.01f);
    float mm  = fin ? 1.f : 0.f;
    out_means[t] = ms * mm;
    out_final[t] = mm;
    s_mask[t] = mm;
  }
  __syncthreads();
  if (t < MAXP * NJ) {
    int p = t / NJ;
    float mm = s_mask[p];
    out_poses[t * 3 + 0] = s_px[t] * mm;
    out_poses[t * 3 + 1] = s_py[t] * mm;
    out_poses[t * 3 + 2] = s_score[p] * s_hv[t] * mm;
  }
}

extern "C" void kernel_launch(void* const* d_in, const int* in_sizes, int n_in,
                              void* d_out, int out_size, void* d_ws, size_t ws_size,
                              hipStream_t stream) {
  const float* heat = (const float*)d_in[0];   // (1,18,256,256) f32
  const float* offs = (const float*)d_in[1];   // (1,34,256,256) f32
  float* out  = (float*)d_out;                 // poses|means|final|heatmap_sum
  float* peak = (float*)d_ws;                  // 65536 floats scratch

  dekr_upsample<<<(NCH * 256 * 256) / 256, 256, 0, stream>>>(heat, out + OFF_HSUM);
  dekr_peak<<<65536 / 256, 256, 0, stream>>>(heat, peak);
  dekr_decode<<<1, 1024, 0, stream>>>(offs, peak, out + OFF_HSUM,
                                      out + OFF_POSES, out + OFF_MEANS,
                                      out + OFF_FINAL);
}